// MessageBuildingLayerLSH_75900662054925
// MI455X (gfx1250) — compile-verified
//
#include <hip/hip_runtime.h>
#include <hip/hip_bf16.h>
#include <math.h>

// ---------------------------------------------------------------------------
// MessageBuildingLayerLSH for MI455X (gfx1250, wave32)
// B=8, N=16384, D_MSG=128, D_NODE=256, n_bins=128, BIN_SIZE=128
// Bandwidth-bound (~360 MB traffic, ~6.5 GFLOP): fp32 WMMA 16x16x4 for the
// GEMM-shaped parts, TDM gather-mode tensor_load_to_lds for the per-bin tile
// gather, nontemporal streaming for write-once outputs.
// ---------------------------------------------------------------------------

#define BATCH   8
#define NPTS    16384
#define DMSG    128
#define DNODE   256
#define NBINS   128
#define BINSZ   128
#define NCOLS   64          // n_bins // 2
#define CBSTRIDE 100        // MAX_NUM_BINS // 2 (codebook row stride)
#define XSTRIDE 132         // padded LDS row stride (matches TDM pad 128+4 dwords)

typedef __attribute__((ext_vector_type(2))) float        v2f;
typedef __attribute__((ext_vector_type(4))) float        v4f;
typedef __attribute__((ext_vector_type(8))) float        v8f;
typedef __attribute__((ext_vector_type(4))) unsigned int v4u;
typedef __attribute__((ext_vector_type(4))) int          v4i;
typedef __attribute__((ext_vector_type(8))) int          v8i;

#if defined(__has_builtin)
#  if __has_builtin(__builtin_amdgcn_tensor_load_to_lds)
#    define USE_TDM 1
#  endif
#endif
#ifndef USE_TDM
#  define USE_TDM 0
#endif

__device__ __forceinline__ void wait_tensorcnt0() {
#if defined(__has_builtin) && __has_builtin(__builtin_amdgcn_s_wait_tensorcnt)
    __builtin_amdgcn_s_wait_tensorcnt(0);
#else
    asm volatile("s_wait_tensorcnt 0x0" ::: "memory");
#endif
}

// ---------------------------------------------------------------------------
// Kernel A: LSH projection (16 pts x 64 cols, K=128) via V_WMMA_F32_16X16X4_F32
// then per-row argmax over symmetric codes [mul, -mul]; +127 for masked pts.
// One wave (32 threads) per 16 points. grid = B * N/16 = 8192 blocks.
// ---------------------------------------------------------------------------
__global__ __launch_bounds__(32)
void lsh_argmax_kernel(const float* __restrict__ x_msg,
                       const float* __restrict__ codebook,
                       const unsigned char* __restrict__ msk,
                       int* __restrict__ bin_idx) {
    __shared__ float lds_mul[16][NCOLS];

    const int blk  = blockIdx.x;
    const int b    = blk >> 10;           // / (N/16)
    const int tile = blk & 1023;
    const int row0 = tile << 4;           // first point of this 16-row tile

    const int lane = threadIdx.x;
    const int m    = lane & 15;           // row (A) / col (B) within tile
    const int hi   = lane >> 4;           // 0: K={0,1}, 1: K={2,3}

    const float* Abase = x_msg + ((size_t)b * NPTS + row0) * DMSG;

    v8f acc[4];
    #pragma unroll
    for (int t = 0; t < 4; ++t) acc[t] = (v8f){0.f,0.f,0.f,0.f,0.f,0.f,0.f,0.f};

    for (int kk = 0; kk < DMSG / 4; ++kk) {
        const int kb = 4 * kk + 2 * hi;
        // A fragment: 16x4 f32 (lane = row M, VGPR pair = K,K+1; lanes 16-31 K+2,K+3)
        v2f a;
        a.x = Abase[(size_t)m * DMSG + kb];
        a.y = Abase[(size_t)m * DMSG + kb + 1];
        #pragma unroll
        for (int t = 0; t < 4; ++t) {
            // B fragment: 4x16 f32, B[k][n] = codebook[k][t*16 + n]
            v2f bf;
            bf.x = codebook[(size_t)kb * CBSTRIDE + t * 16 + m];
            bf.y = codebook[(size_t)(kb + 1) * CBSTRIDE + t * 16 + m];
            acc[t] = __builtin_amdgcn_wmma_f32_16x16x4_f32(
                false, a, false, bf, (short)0, acc[t], false, false);
        }
    }

    // C layout: lanes 0-15 -> M = vgpr, lanes 16-31 -> M = vgpr+8; N = lane%16
    #pragma unroll
    for (int t = 0; t < 4; ++t)
        #pragma unroll
        for (int v = 0; v < 8; ++v)
            lds_mul[v + 8 * hi][t * 16 + m] = acc[t][v];

    __syncthreads();

    if (lane < 16) {
        const int row = lane;
        float best = -INFINITY;
        int   idx  = 0;
        // strict '>' preserves jnp.argmax first-occurrence tie rule
        #pragma unroll 4
        for (int j = 0; j < NCOLS; ++j) {
            float v = lds_mul[row][j];
            if (v > best) { best = v; idx = j; }
        }
        #pragma unroll 4
        for (int j = 0; j < NCOLS; ++j) {
            float v = -lds_mul[row][j];
            if (v > best) { best = v; idx = NCOLS + j; }
        }
        const int n = row0 + row;
        if (!msk[(size_t)b * NPTS + n]) idx += NBINS - 1;
        bin_idx[(size_t)b * NPTS + n] = idx;
    }
}

// ---------------------------------------------------------------------------
// Kernel B: stable counting sort of 16384 keys (0..254) per batch.
// One block (256 threads) per batch; matches jnp.argsort stable semantics.
// ---------------------------------------------------------------------------
__global__ __launch_bounds__(256)
void stable_sort_kernel(const int* __restrict__ bin_idx,
                        int* __restrict__ bins_int,
                        float* __restrict__ bins_out_f) {
    __shared__ int hist[256];   // running exclusive offsets
    __shared__ int keys[256];
    __shared__ int chist[256];

    const int b   = blockIdx.x;
    const int tid = threadIdx.x;
    const int* kin = bin_idx + (size_t)b * NPTS;

    hist[tid] = 0;
    __syncthreads();
    for (int n = tid; n < NPTS; n += 256)
        atomicAdd(&hist[kin[n]], 1);
    __syncthreads();

    if (tid == 0) {           // serial exclusive prefix over 256 entries
        int run = 0;
        for (int k = 0; k < 256; ++k) { int c = hist[k]; hist[k] = run; run += c; }
    }
    __syncthreads();

    for (int c = 0; c < NPTS / 256; ++c) {
        const int n   = c * 256 + tid;
        const int key = kin[n];
        keys[tid]  = key;
        chist[tid] = 0;
        __syncthreads();

        int rank = 0;                       // stable rank within chunk
        for (int t = 0; t < tid; ++t) rank += (keys[t] == key);
        const int pos = hist[key] + rank;
        bins_int[(size_t)b * NPTS + pos]   = n;
        bins_out_f[(size_t)b * NPTS + pos] = (float)n;
        atomicAdd(&chist[key], 1);
        __syncthreads();

        hist[tid] += chist[tid];
        __syncthreads();
    }
}

// ---------------------------------------------------------------------------
// Kernel C1: gather x_node rows into bins, float4-coalesced, nontemporal
// (128 MB each way, zero reuse -> keep it out of L2).
// ---------------------------------------------------------------------------
__global__ __launch_bounds__(256)
void gather_features_kernel(const float* __restrict__ x_node,
                            const int* __restrict__ bins_int,
                            float* __restrict__ out_feat) {
    const long long tid = (long long)blockIdx.x * 256 + threadIdx.x;   // over B*N*64 float4s
    const int d4     = (int)(tid & 63);             // float4 index within row
    const int p_glob = (int)(tid >> 6);             // b*N + p
    const int b      = p_glob >> 14;
    const int n      = bins_int[p_glob];
    if (d4 == 0 && p_glob + 512 < BATCH * NPTS)     // global_prefetch_b8 of index stream
        __builtin_prefetch(&bins_int[p_glob + 512], 0, 1);
    const v4f* src = (const v4f*)x_node + ((size_t)b * NPTS + n) * (DNODE / 4) + d4;
    v4f v = __builtin_nontemporal_load(src);
    __builtin_nontemporal_store(v, (v4f*)out_feat + tid);
}

// ---------------------------------------------------------------------------
// Kernel C2: gather mask into bins as float.
// ---------------------------------------------------------------------------
__global__ __launch_bounds__(256)
void gather_msk_kernel(const unsigned char* __restrict__ msk,
                       const int* __restrict__ bins_int,
                       float* __restrict__ out_msk) {
    const int tid = blockIdx.x * 256 + threadIdx.x;   // over B*N
    const int b   = tid >> 14;
    const int n   = bins_int[tid];
    out_msk[tid] = msk[(size_t)b * NPTS + n] ? 1.0f : 0.0f;
}

// ---------------------------------------------------------------------------
// Kernel D: per-bin Gaussian kernel matrix.
// One block (8 waves) per (batch, bin). The 128x128 f32 tile is gathered into
// 68 KB dynamic LDS by the Tensor Data Mover in gather mode: each wave issues
// one tensor_load_to_lds descriptor moving 16 indexed rows (16-bit indices),
// with TDM padding (128 dwords data + 4 dwords skip) producing the 132-float
// LDS row stride. Masking is applied only in the epilogue (equivalent: every
// dm element touching a masked row is multiplied by zero anyway).
// Gram = X * X^T via 64 WMMA f32_16x16x4 tiles; fused sqrt/exp/mask epilogue,
// nontemporal coalesced stores (67 MB write-once output).
// ---------------------------------------------------------------------------
__global__ __launch_bounds__(256)
void gram_dm_kernel(const float* __restrict__ x_msg,
                    const unsigned char* __restrict__ msk,
                    const int* __restrict__ bins_int,
                    float* __restrict__ dm_out) {
    extern __shared__ __align__(16) char smem[];
    float* X    = (float*)smem;                         // [128][XSTRIDE]
    float* sq   = (float*)(smem + BINSZ * XSTRIDE * 4); // [128]
    float* mrow = sq + BINSZ;                           // [128]

    const int blk  = blockIdx.x;        // b*NBINS + bin
    const int b    = blk >> 7;
    const int wave = threadIdx.x >> 5;
    const int lane = threadIdx.x & 31;

#if USE_TDM
    // ---- Phase 1: TDM gather. One descriptor per wave = 16 indexed rows. ----
    {
        const int rbase = blk * BINSZ + wave * 16;
        unsigned idx[16];
        #pragma unroll
        for (int k = 0; k < 16; ++k)
            idx[k] = (unsigned)bins_int[rbase + k] & 0xffffu;

        const unsigned long long gaddr =
            (unsigned long long)(const void*)(x_msg + (size_t)b * NPTS * DMSG);
        // Low 32 bits of a flat shared-aperture pointer are the LDS byte address.
        const unsigned lds_addr =
            (unsigned)(unsigned long long)(void*)X + (unsigned)(wave * 16 * XSTRIDE * 4);

        // D# group 0: count=1, gather_mode=1, 16-bit indices, type=2
        v4u g0;
        g0.x = 1u | (1u << 31);
        g0.y = lds_addr;
        g0.z = (unsigned)gaddr;
        g0.w = ((unsigned)(gaddr >> 32) & 0x1FFFFFFu) | (2u << 30);

        // D# group 1: data_size=4B, pad_enable, pad_interval=6 (128 dw),
        // pad_amount=3 (4 dw); tensor 128 x 16384, tile_dim0=128, 16 indices.
        v8i g1;
        g1[0] = (2 << 16) | (1 << 20) | (6 << 22) | (3 << 25);
        g1[1] = (DMSG & 0xffff) << 16;          // tensor_dim0 = 128
        g1[2] = (NPTS & 0xffff) << 16;          // tensor_dim1 = 16384
        g1[3] = (DMSG & 0xffff) << 16;          // tile_dim0 = 128
        g1[4] = 16;                             // tile_dim1 = #valid indices
        g1[5] = DMSG;                           // tensor_dim0_stride = 128
        g1[6] = 0;
        g1[7] = 0;

        // D# groups 2/3: sixteen 16-bit row indices
        v4i g2, g3;
        #pragma unroll
        for (int k = 0; k < 4; ++k) {
            g2[k] = (int)(idx[2 * k]     | (idx[2 * k + 1] << 16));
            g3[k] = (int)(idx[8 + 2 * k] | (idx[9 + 2 * k] << 16));
        }

#if defined(__clang_major__) && (__clang_major__ >= 23)
        __builtin_amdgcn_tensor_load_to_lds(g0, g1, g2, g3, (v8i)0, 0);
#else
        __builtin_amdgcn_tensor_load_to_lds(g0, g1, g2, g3, 0);
#endif
    }

    // Overlap: row masks while the TDM is in flight.
    if (threadIdx.x < BINSZ) {
        const int n = bins_int[(size_t)blk * BINSZ + threadIdx.x];
        mrow[threadIdx.x] = msk[(size_t)b * NPTS + n] ? 1.0f : 0.0f;
    }
    wait_tensorcnt0();
    __syncthreads();
#else
    // ---- Fallback phase 1: VMEM gather (unmasked; mask applied in epilogue).
    for (int r = wave; r < BINSZ; r += 8) {
        const int n = bins_int[(size_t)blk * BINSZ + r];
        if (lane == 0) mrow[r] = msk[(size_t)b * NPTS + n] ? 1.0f : 0.0f;
        const v4f v = *((const v4f*)(x_msg + ((size_t)b * NPTS + n) * DMSG) + lane);
        *(v4f*)&X[r * XSTRIDE + 4 * lane] = v;
    }
    __syncthreads();
#endif

    // Phase 2: squared norms
    if (threadIdx.x < BINSZ) {
        float s = 0.f;
        const float* row = &X[threadIdx.x * XSTRIDE];
        #pragma unroll 8
        for (int d = 0; d < DMSG; ++d) s += row[d] * row[d];
        sq[threadIdx.x] = s;
    }
    __syncthreads();

    // Phase 3: gram tiles + fused epilogue
    const int m  = lane & 15;
    const int hi = lane >> 4;
    float* outb = dm_out + (size_t)blk * BINSZ * BINSZ;

    #pragma unroll 1
    for (int t8 = 0; t8 < 8; ++t8) {
        const int tile = wave + 8 * t8;      // 0..63, uniform per wave
        const int i0   = (tile >> 3) << 4;
        const int j0   = (tile & 7) << 4;

        v8f c = (v8f){0.f,0.f,0.f,0.f,0.f,0.f,0.f,0.f};
        for (int kk = 0; kk < DMSG / 4; ++kk) {
            const int kb = 4 * kk + 2 * hi;
            v2f a, bf;
            a.x  = X[(i0 + m) * XSTRIDE + kb];      // A[M][K] = X[i0+M][K]
            a.y  = X[(i0 + m) * XSTRIDE + kb + 1];
            bf.x = X[(j0 + m) * XSTRIDE + kb];      // B[K][N] = X[j0+N][K]
            bf.y = X[(j0 + m) * XSTRIDE + kb + 1];
            c = __builtin_amdgcn_wmma_f32_16x16x4_f32(
                false, a, false, bf, (short)0, c, false, false);
        }

        // Epilogue: element (i, j) = (i0 + v + 8*hi, j0 + lane%16)
        const int j = j0 + m;
        const float sqj = sq[j];
        const float mj  = mrow[j];
        #pragma unroll
        for (int v = 0; v < 8; ++v) {
            const int i = i0 + v + 8 * hi;
            const float d2   = sq[i] - 2.0f * c[v] + sqj;
            const float dist = sqrtf(fmaxf(d2, 1e-6f));
            float dmv = expf(-0.1f * dist);
            dmv = fminf(dmv, 1.0f) * mrow[i] * mj;
            __builtin_nontemporal_store(dmv, &outb[(size_t)i * BINSZ + j]);
        }
    }
}

// ---------------------------------------------------------------------------
// Host-side launch. Outputs concatenated flat (float32):
//   bins_split        [B,128,128]      ->       131072 @ 0
//   x_features_binned [B,128,128,256]  ->     33554432 @ 131072
//   dm                [B,128,128,128,1]->     16777216 @ 33685504
//   msk_f_binned      [B,128,128,1]    ->       131072 @ 50462720
// ---------------------------------------------------------------------------
extern "C" void kernel_launch(void* const* d_in, const int* in_sizes, int n_in,
                              void* d_out, int out_size, void* d_ws, size_t ws_size,
                              hipStream_t stream) {
    const float*         x_msg    = (const float*)d_in[0];
    const float*         x_node   = (const float*)d_in[1];
    const float*         codebook = (const float*)d_in[2];
    const unsigned char* msk      = (const unsigned char*)d_in[3];

    float* out        = (float*)d_out;
    float* bins_f     = out;
    float* feat_out   = out + 131072;
    float* dm_out     = out + 33685504;
    float* mskf_out   = out + 50462720;

    int* bin_idx  = (int*)d_ws;                 //  512 KB
    int* bins_int = bin_idx + BATCH * NPTS;     //  512 KB

    // A: LSH projection + argmax -> bin keys
    lsh_argmax_kernel<<<dim3(BATCH * NPTS / 16), dim3(32), 0, stream>>>(
        x_msg, codebook, msk, bin_idx);

    // B: stable counting sort per batch -> permutation
    stable_sort_kernel<<<dim3(BATCH), dim3(256), 0, stream>>>(
        bin_idx, bins_int, bins_f);

    // C1: gather node features (float4 coalesced, nontemporal)
    gather_features_kernel<<<dim3(BATCH * NPTS * (DNODE / 4) / 256), dim3(256), 0, stream>>>(
        x_node, bins_int, feat_out);

    // C2: gather mask
    gather_msk_kernel<<<dim3(BATCH * NPTS / 256), dim3(256), 0, stream>>>(
        msk, bins_int, mskf_out);

    // D: per-bin gram + Gaussian kernel, 68 KB dynamic LDS per workgroup
    const size_t shmem = (size_t)BINSZ * XSTRIDE * sizeof(float) + 2 * BINSZ * sizeof(float);
    gram_dm_kernel<<<dim3(BATCH * NBINS), dim3(256), shmem, stream>>>(
        x_msg, msk, bins_int, dm_out);
}